// TransportMap_38895223833112
// MI455X (gfx1250) — compile-verified
//
#include <hip/hip_runtime.h>
#include <math.h>

typedef __attribute__((ext_vector_type(2))) float v2f;
typedef __attribute__((ext_vector_type(8))) float v8f;

#define N_PTS   128     // n
#define N_BATCH 2048    // N
#define P_DIM   2
#define M_MAX   64
#define KD      32      // m + p (padded feature count)
#define WS      36      // W row stride (bank-conflict free)
#define GS      130     // G row stride
#define SQRT3F  1.7320508075688772f

__global__ __launch_bounds__(256)
void TransportMap_kernel(const float* __restrict__ Y,
                         const float* __restrict__ X,
                         const float* __restrict__ theta,
                         const int*   __restrict__ NNmax,
                         const int*   __restrict__ mptr,
                         float*       __restrict__ out)
{
    __shared__ float sW[N_PTS * WS];    // 18 KB: feature matrix W (128 x 32)
    __shared__ float sG[N_PTS * GS];    // 65 KB: G, then its Cholesky factor
    __shared__ float sSq[N_PTS];
    __shared__ float sY[N_PTS];
    __shared__ float sYs[N_PTS];
    __shared__ int   sXz;

    const int b    = blockIdx.x;
    const int tid  = threadIdx.x;
    const int lane = tid & 31;
    const int wave = tid >> 5;
    const int lhi  = lane >> 4;     // 0/1 : K-half for A/B frags, M-half for C/D
    const int llo  = lane & 15;

    float th[10];
    #pragma unroll
    for (int k = 0; k < 10; ++k) th[k] = theta[k];
    int m = *mptr; if (m > KD - P_DIM) m = KD - P_DIM;

    const float logs   = -logf((float)(b + 1));
    float nug = expf(logs * th[1] + th[0]) - 1e-5f;
    nug = (nug > 0.0f ? nug : 0.0f) + 1e-5f;
    const float alpha  = 1.0f / (4.0f * 4.0f) + 2.0f;      // NUG_MULT = 4
    const float beta   = nug * (alpha - 1.0f);
    const float sigma  = expf(logs * th[4] + th[3]);
    const float sig2   = sigma * sigma;
    const float lenScal = expf(th[5]) * sqrtf(2.0f * 1.5f); // SMOOTH = 1.5
    const float invl2  = 1.0f / (lenScal * lenScal);
    const float invNug = 1.0f / nug;

    // ---- xz = all(X[:, b, :] == 0)  (n*p = 256 values, one per thread) ----
    if (tid == 0) sXz = 0;
    __syncthreads();
    {
        int i = tid >> 1, j = tid & 1;
        float xv = X[((size_t)i * N_BATCH + b) * P_DIM + j];
        if (xv != 0.0f) atomicOr(&sXz, 1);
    }
    __syncthreads();
    const bool xz = (sXz == 0);

    // ---- build W = [Ys | Xs] in LDS ----
    const int ncol = (b < m) ? b : m;
    for (int idx = tid; idx < N_PTS * KD; idx += 256) {
        int i = idx >> 5;
        int j = idx & 31;
        float w = 0.0f;
        if (j < m) {
            if (j < ncol) {
                int nn = NNmax[b * M_MAX + j];
                float scol = sqrtf(expf((float)(j + 1) * th[2]));
                w = Y[(size_t)i * N_BATCH + nn] * scol;
            }
        } else if (j < m + P_DIM) {
            int jx = j - m;
            if (!xz) {
                float sx = expf(logs * th[6 + P_DIM + jx] + th[6 + jx]);
                w = X[((size_t)i * N_BATCH + b) * P_DIM + jx] * sx;
            }
        }
        sW[i * WS + j] = w;
    }
    if (tid < N_PTS) sY[tid] = Y[(size_t)tid * N_BATCH + b];
    __syncthreads();
    if (tid < N_PTS) {
        float s = 0.0f;
        #pragma unroll
        for (int k = 0; k < KD; ++k) { float w = sW[tid * WS + k]; s += w * w; }
        sSq[tid] = s;
    }
    __syncthreads();

    // ---- SYRK lin = W * W^T via V_WMMA_F32_16X16X4_F32, fused Matern -> G ----
    const int rb = wave;                    // each wave: one 16-row block
    for (int cb = 0; cb < 8; ++cb) {
        v8f acc = {};
        #pragma unroll
        for (int kk = 0; kk < KD; kk += 4) {
            v2f a, bb;
            int ka = kk + 2 * lhi;          // lanes 0-15: K={kk,kk+1}; 16-31: +2
            a.x  = sW[(rb * 16 + llo) * WS + ka];
            a.y  = sW[(rb * 16 + llo) * WS + ka + 1];
            bb.x = sW[(cb * 16 + llo) * WS + ka];
            bb.y = sW[(cb * 16 + llo) * WS + ka + 1];
            acc = __builtin_amdgcn_wmma_f32_16x16x4_f32(
                      false, a, false, bb, (short)0, acc, false, false);
        }
        #pragma unroll
        for (int r = 0; r < 8; ++r) {
            int i = rb * 16 + r + 8 * lhi;
            int j = cb * 16 + llo;
            float lin = acc[r];
            float r2 = (sSq[i] + sSq[j] - 2.0f * lin) * invl2;
            r2 = fmaxf(r2, 0.0f);
            float c = SQRT3F * sqrtf(r2);
            float matern = (1.0f + c) * expf(-c);
            float Kv = (lin + matern * sig2) * invNug;
            if (b == 0) Kv = 0.0f;
            sG[i * GS + j] = Kv + ((i == j) ? 1.0f : 0.0f);
        }
    }
    __syncthreads();

    // ---- blocked Cholesky of G (lower), 16x16 panels, WMMA trailing updates ----
    volatile float* vG = sG;
    for (int kb = 0; kb < 8; ++kb) {
        const int k0 = kb * 16;

        // diagonal block: wave 0, lane i = row, right-looking
        if (wave == 0 && lane < 16) {
            int i = lane;
            for (int j = 0; j < 16; ++j) {
                float l = sqrtf(vG[(k0 + j) * GS + k0 + j]);
                if (i == j) vG[(k0 + j) * GS + k0 + j] = l;
                if (i > j) {
                    float v = vG[(k0 + i) * GS + k0 + j] / l;
                    vG[(k0 + i) * GS + k0 + j] = v;
                    for (int t = j + 1; t <= i; ++t)
                        vG[(k0 + i) * GS + k0 + t] -= v * vG[(k0 + t) * GS + k0 + j];
                }
            }
        }
        __syncthreads();

        // panel solve: rows below, one thread per row: x * Ldiag^T = A_row
        const int nrows = N_PTS - k0 - 16;
        for (int rr = tid; rr < nrows; rr += 256) {
            int i = k0 + 16 + rr;
            float x[16];
            #pragma unroll
            for (int j = 0; j < 16; ++j) {
                float s = sG[i * GS + k0 + j];
                for (int t = 0; t < j; ++t) s -= x[t] * sG[(k0 + j) * GS + k0 + t];
                x[j] = s / sG[(k0 + j) * GS + k0 + j];
            }
            #pragma unroll
            for (int j = 0; j < 16; ++j) sG[i * GS + k0 + j] = x[j];
        }
        __syncthreads();

        // trailing update: G[ib,jb] -= Lp[ib] * Lp[jb]^T  (WMMA, K=16)
        const int nb = 7 - kb;
        const int ntiles = nb * (nb + 1) / 2;
        for (int t = wave; t < ntiles; t += 8) {
            int r = 0;
            while ((r + 1) * (r + 2) / 2 <= t) ++r;
            int ib = kb + 1 + r;
            int jb = kb + 1 + (t - r * (r + 1) / 2);
            v8f acc = {};
            #pragma unroll
            for (int kk = 0; kk < 16; kk += 4) {
                v2f a, bb;
                int ka = k0 + kk + 2 * lhi;
                a.x  = sG[(ib * 16 + llo) * GS + ka];
                a.y  = sG[(ib * 16 + llo) * GS + ka + 1];
                bb.x = sG[(jb * 16 + llo) * GS + ka];
                bb.y = sG[(jb * 16 + llo) * GS + ka + 1];
                acc = __builtin_amdgcn_wmma_f32_16x16x4_f32(
                          false, a, false, bb, (short)0, acc, false, false);
            }
            #pragma unroll
            for (int rq = 0; rq < 8; ++rq) {
                int i = ib * 16 + rq + 8 * lhi;
                int j = jb * 16 + llo;
                sG[i * GS + j] -= acc[rq];
            }
        }
        __syncthreads();
    }

    // ---- forward substitution L y = Y[:, b]  (wave 0, shuffle-reduced dots) ----
    volatile float* vYs = sYs;
    if (wave == 0) {
        for (int i = 0; i < N_PTS; ++i) {
            float s = 0.0f;
            for (int t = lane; t < i; t += 32) s += sG[i * GS + t] * vYs[t];
            #pragma unroll
            for (int off = 16; off > 0; off >>= 1) s += __shfl_xor(s, off, 32);
            if (lane == 0) vYs[i] = (sY[i] - s) / sG[i * GS + i];
        }
    }
    __syncthreads();

    // ---- log-likelihood reduction ----
    if (wave == 0) {
        float ss = 0.0f, ld = 0.0f;
        for (int i = lane; i < N_PTS; i += 32) {
            float yv = sYs[i];
            ss += yv * yv;
            ld += logf(sG[i * GS + i]);
        }
        #pragma unroll
        for (int off = 16; off > 0; off >>= 1) {
            ss += __shfl_xor(ss, off, 32);
            ld += __shfl_xor(ld, off, 32);
        }
        if (lane == 0) {
            float alphaPost = alpha + (float)N_PTS * 0.5f;
            float betaPost  = beta + 0.5f * ss;
            float loglik = -ld + alpha * logf(beta) - alphaPost * logf(betaPost)
                         + lgammaf(alphaPost) - lgammaf(alpha);
            atomicAdd(out, -loglik);
        }
    }
}

__global__ void zero_out_kernel(float* out, int nel) {
    int i = blockIdx.x * blockDim.x + threadIdx.x;
    if (i < nel) out[i] = 0.0f;
}

extern "C" void kernel_launch(void* const* d_in, const int* in_sizes, int n_in,
                              void* d_out, int out_size, void* d_ws, size_t ws_size,
                              hipStream_t stream) {
    (void)in_sizes; (void)n_in; (void)d_ws; (void)ws_size;
    const float* Y     = (const float*)d_in[0];
    const float* X     = (const float*)d_in[1];
    const float* theta = (const float*)d_in[2];
    const int*   NN    = (const int*)d_in[3];
    const int*   m     = (const int*)d_in[4];
    float* out = (float*)d_out;

    zero_out_kernel<<<1, 32, 0, stream>>>(out, out_size);
    TransportMap_kernel<<<N_BATCH, 256, 0, stream>>>(Y, X, theta, NN, m, out);
}